// AdaptiveFusion_19765439496515
// MI455X (gfx1250) — compile-verified
//
#include <hip/hip_runtime.h>
#include <math.h>

// ---------------------------------------------------------------------------
// Fused gate + residual-mix + LayerNorm, bandwidth-bound streaming kernel.
// B*T = 32768 rows, D = 1024, fp32 in / fp32 out. ~403 MB touched once
// => non-temporal on both streams (L2 is 192 MB; zero reuse).
// Data path: global --(GLOBAL_LOAD_ASYNC_TO_LDS_B128 th:NT, ASYNCcnt)-->
//            LDS --> VGPR --> global_store_b128 th:NT.
// ---------------------------------------------------------------------------

#define DDIM 1024
#define TPB  256            // 8 wave32 waves
#define VPT  4              // floats per thread: DDIM / TPB
#define PIPE 3              // async LDS pipeline depth
#define RPB  16             // rows per block

typedef float v4f __attribute__((ext_vector_type(4)));  // native vec for NT store

// Low 32 bits of a generic pointer to __shared__ == LDS byte offset (AS3).
__device__ __forceinline__ unsigned lds_off(const void* p) {
    return (unsigned)(unsigned long long)p;
}

// Async DMA: copies 16 bytes per lane from global (saddr + voff) to the LDS address in lds_addr.
// Non-temporal: the stream is consumed exactly once (real home is LDS).
__device__ __forceinline__ void async_load_b128_nt(unsigned lds_addr,
                                                   const float* sbase,
                                                   int voff_bytes) {
    asm volatile("global_load_async_to_lds_b128 %0, %1, %2 th:TH_LOAD_NT"
                 :: "v"(lds_addr), "v"(voff_bytes), "s"(sbase)
                 : "memory");
}

__device__ __forceinline__ void s_wait_asynccnt_4() {
    asm volatile("s_wait_asynccnt 0x4" ::: "memory");
}

__global__ __launch_bounds__(TPB)
void AdaptiveFusion_kernel(const float* __restrict__ x1,
                           const float* __restrict__ x2,
                           const float* __restrict__ W1,
                           const float* __restrict__ W2,
                           const float* __restrict__ gamma,
                           const float* __restrict__ beta,
                           float* __restrict__ out,
                           int nrows) {
    __shared__ float buf[PIPE][2][DDIM];   // [slot][input][col]  (24 KB)
    __shared__ float red4[8][4];
    __shared__ float red2[8][2];

    const int tid  = threadIdx.x;
    const int wid  = tid >> 5;
    const int lane = tid & 31;
    const int col  = tid * VPT;
    const int thr_byte = col * 4;

    // Per-thread constants for this column range (reused by every block ->
    // keep regular-temporal so they stay resident in L2).
    const float4 w1a = *(const float4*)(W1 + col);          // W1[0, col..]
    const float4 w1b = *(const float4*)(W1 + DDIM + col);   // W1[1, col..]
    const float4 w2a = *(const float4*)(W2 + col);
    const float4 w2b = *(const float4*)(W2 + DDIM + col);
    const float4 gv  = *(const float4*)(gamma + col);
    const float4 bv  = *(const float4*)(beta + col);

    const int r0 = blockIdx.x * RPB;
    if (r0 >= nrows) return;
    int rend = r0 + RPB; if (rend > nrows) rend = nrows;
    const int nr = rend - r0;

    // ---- prologue: prefetch rows 0..PIPE-2 (clamped) ----
    #pragma unroll
    for (int p = 0; p < PIPE - 1; ++p) {
        int rr = p; if (rr > nr - 1) rr = nr - 1;
        const int goff = (r0 + rr) * (DDIM * 4) + thr_byte;
        async_load_b128_nt(lds_off(&buf[p][0][col]), x1, goff);
        async_load_b128_nt(lds_off(&buf[p][1][col]), x2, goff);
    }

    for (int i = 0; i < nr; ++i) {
        // Issue prefetch of row i+PIPE-1 (clamped so exactly 2 async ops
        // are always issued per stage -> wait<=4 guarantees row i landed).
        {
            int pr = i + (PIPE - 1); if (pr > nr - 1) pr = nr - 1;
            const int pslot = (i + (PIPE - 1)) % PIPE;
            const int goff = (r0 + pr) * (DDIM * 4) + thr_byte;
            async_load_b128_nt(lds_off(&buf[pslot][0][col]), x1, goff);
            async_load_b128_nt(lds_off(&buf[pslot][1][col]), x2, goff);
        }

        s_wait_asynccnt_4();     // own slices of row i landed in LDS
        __syncthreads();         // all 8 waves' slices visible

        const int s = i % PIPE;
        const float4 a = *(const float4*)&buf[s][0][col];
        const float4 b = *(const float4*)&buf[s][1][col];

        // ---- gate logits: 4 partial dots, block reduce ----
        float p0 = a.x*w1a.x + a.y*w1a.y + a.z*w1a.z + a.w*w1a.w;
        float p1 = a.x*w1b.x + a.y*w1b.y + a.z*w1b.z + a.w*w1b.w;
        float p2 = b.x*w2a.x + b.y*w2a.y + b.z*w2a.z + b.w*w2a.w;
        float p3 = b.x*w2b.x + b.y*w2b.y + b.z*w2b.z + b.w*w2b.w;
        #pragma unroll
        for (int off = 16; off; off >>= 1) {
            p0 += __shfl_xor(p0, off);
            p1 += __shfl_xor(p1, off);
            p2 += __shfl_xor(p2, off);
            p3 += __shfl_xor(p3, off);
        }
        if (lane == 0) {
            red4[wid][0] = p0; red4[wid][1] = p1;
            red4[wid][2] = p2; red4[wid][3] = p3;
        }
        __syncthreads();
        float z1 = 0.f, z2 = 0.f;
        #pragma unroll
        for (int w = 0; w < 8; ++w) {
            z1 += red4[w][0] + red4[w][2];
            z2 += red4[w][1] + red4[w][3];
        }
        const float lam1 = 1.f / (1.f + __expf(-z1));
        const float lam2 = 1.f / (1.f + __expf(-z2));
        const float c1 = 1.f + lam1, c2 = 1.f + lam2;

        // ---- fused = (1+lam1)*x1 + (1+lam2)*x2 ----
        float4 f;
        f.x = c1*a.x + c2*b.x;
        f.y = c1*a.y + c2*b.y;
        f.z = c1*a.z + c2*b.z;
        f.w = c1*a.w + c2*b.w;

        // ---- LayerNorm reduction: sum, sum of squares ----
        float q0 = f.x + f.y + f.z + f.w;
        float q1 = f.x*f.x + f.y*f.y + f.z*f.z + f.w*f.w;
        #pragma unroll
        for (int off = 16; off; off >>= 1) {
            q0 += __shfl_xor(q0, off);
            q1 += __shfl_xor(q1, off);
        }
        if (lane == 0) { red2[wid][0] = q0; red2[wid][1] = q1; }
        __syncthreads();
        float S = 0.f, Q = 0.f;
        #pragma unroll
        for (int w = 0; w < 8; ++w) { S += red2[w][0]; Q += red2[w][1]; }

        const float mean = S * (1.f / DDIM);
        const float var  = Q * (1.f / DDIM) - mean * mean;
        const float inv  = rsqrtf(var + 1e-5f);

        // Write-once stream: non-temporal 128-bit store (th:TH_STORE_NT).
        v4f o;
        o.x = (f.x - mean) * inv * gv.x + bv.x;
        o.y = (f.y - mean) * inv * gv.y + bv.y;
        o.z = (f.z - mean) * inv * gv.z + bv.z;
        o.w = (f.w - mean) * inv * gv.w + bv.w;
        __builtin_nontemporal_store(o, (v4f*)(out + (size_t)(r0 + i) * DDIM + col));
    }
}

extern "C" void kernel_launch(void* const* d_in, const int* in_sizes, int n_in,
                              void* d_out, int out_size, void* d_ws, size_t ws_size,
                              hipStream_t stream) {
    (void)n_in; (void)out_size; (void)d_ws; (void)ws_size;
    const float* x1 = (const float*)d_in[0];
    const float* x2 = (const float*)d_in[1];
    const float* W1 = (const float*)d_in[2];
    const float* W2 = (const float*)d_in[3];
    const float* g  = (const float*)d_in[4];
    const float* b  = (const float*)d_in[5];
    float* out = (float*)d_out;

    const int nrows = in_sizes[0] / DDIM;            // B*T = 32768
    const int grid  = (nrows + RPB - 1) / RPB;       // 2048 blocks
    hipLaunchKernelGGL(AdaptiveFusion_kernel, dim3(grid), dim3(TPB), 0, stream,
                       x1, x2, W1, W2, g, b, out, nrows);
}